// GIN_13993003450905
// MI455X (gfx1250) — compile-verified
//
#include <hip/hip_runtime.h>

typedef __attribute__((ext_vector_type(2))) float v2f;
typedef __attribute__((ext_vector_type(8))) float v8f;

constexpr int DF = 64;          // feature dim
constexpr int TILE_ROWS = 128;  // nodes per block in MLP kernel
constexpr int HSTR = 66;        // even padded H-row stride -> aligned b64 A-frag loads
constexpr int WPSTR = 130;      // per-K-pair weight row stride (64 cols * 2 + pad)

__device__ __forceinline__ void fatomic_add(float* p, float v) {
  unsafeAtomicAdd(p, v);        // HW global_atomic_add_f32
}

// ---------------------------------------------------------------- zero fill
__global__ __launch_bounds__(256) void zero_kernel(float* __restrict__ p, long n) {
  long i = (long)blockIdx.x * blockDim.x + threadIdx.x;
  if (i < n) p[i] = 0.0f;
}

// ------------------------------------------------- edge gather + scatter-add
__global__ __launch_bounds__(256) void scatter_kernel(
    const float* __restrict__ x, const int* __restrict__ src,
    const int* __restrict__ dst, float* __restrict__ agg, long total) {
  long i = (long)blockIdx.x * blockDim.x + threadIdx.x;
  if (i >= total) return;
  int e = (int)(i >> 6);
  int d = (int)(i & 63);
  int s = src[e], t = dst[e];
  fatomic_add(&agg[(long)t * DF + d], x[(long)s * DF + d]);
}

// ---------------------------------------------------- fused GIN MLP (WMMA f32)
// h = (1+eps)*x + agg ; t = relu(h@W1 + b1) ; y = t@W2 + b2
// also accumulates per-feature sum / sumsq for BatchNorm.
__global__ __launch_bounds__(256) void gin_mlp_kernel(
    const float* __restrict__ x, const float* __restrict__ agg,
    const float* __restrict__ W1, const float* __restrict__ b1,
    const float* __restrict__ W2, const float* __restrict__ b2,
    const float* __restrict__ eps, int layer,
    float* __restrict__ y, float* __restrict__ gsum, float* __restrict__ gsq,
    int n) {
  __shared__ float sW1[(DF / 2) * WPSTR];   // K-paired: [k/2][col][k&1]
  __shared__ float sW2[(DF / 2) * WPSTR];
  __shared__ float sH[TILE_ROWS * HSTR];

  const int tid = threadIdx.x;
  const long base = (long)blockIdx.x * TILE_ROWS;
  const float e1 = 1.0f + eps[layer];

  // stage weights in LDS, K-paired so a B-fragment is one aligned b64 word
  for (int i = tid; i < DF * DF; i += 256) {
    int k = i >> 6, c = i & 63;
    int d = (k >> 1) * WPSTR + c * 2 + (k & 1);
    sW1[d] = W1[i];
    sW2[d] = W2[i];
  }
  // stage h tile (vectorized: float4 global loads, float2 LDS stores)
  {
    const float4* __restrict__ x4 = (const float4*)x;
    const float4* __restrict__ a4 = (const float4*)agg;
    for (int i = tid; i < TILE_ROWS * (DF / 4); i += 256) {
      int r = i >> 4, c4 = i & 15;
      long gr = base + r;
      float4 h = {0.0f, 0.0f, 0.0f, 0.0f};
      if (gr < n) {
        float4 xv = x4[gr * (DF / 4) + c4];
        float4 av = a4[gr * (DF / 4) + c4];
        h.x = fmaf(e1, xv.x, av.x);
        h.y = fmaf(e1, xv.y, av.y);
        h.z = fmaf(e1, xv.z, av.z);
        h.w = fmaf(e1, xv.w, av.w);
      }
      v2f* p = (v2f*)&sH[r * HSTR + c4 * 4];
      p[0] = (v2f){h.x, h.y};
      p[1] = (v2f){h.z, h.w};
    }
  }
  __syncthreads();

  const int lane = tid & 31;
  const int wave = tid >> 5;   // 8 waves, 16 rows each
  const int m    = lane & 15;  // A-row within tile / C-column within tile
  const int kh   = lane >> 4;  // K-half selector (ISA f32 A/B layout)
  const int arow = 16 * wave + m;

  // ---- GEMM 1: t = h @ W1 + b1 -------------------------------------------
  v8f acc[4];
#pragma unroll
  for (int t = 0; t < 4; ++t) {
    float bv = b1[t * 16 + m];
    acc[t] = (v8f){bv, bv, bv, bv, bv, bv, bv, bv};
  }
  for (int k0 = 0; k0 < DF; k0 += 4) {
    const int ka = k0 + 2 * kh;                       // even
    v2f a = *(const v2f*)&sH[arow * HSTR + ka];       // {h[ka], h[ka+1]} b64
#pragma unroll
    for (int t = 0; t < 4; ++t) {
      v2f b = *(const v2f*)&sW1[(ka >> 1) * WPSTR + (t * 16 + m) * 2];
      acc[t] = __builtin_amdgcn_wmma_f32_16x16x4_f32(
          false, a, false, b, (short)0, acc[t], false, false);
    }
  }

  // ReLU in registers, write intermediate back into this wave's own rows of sH
  __syncthreads();
#pragma unroll
  for (int t = 0; t < 4; ++t) {
#pragma unroll
    for (int j = 0; j < 8; ++j) {
      float v = acc[t][j];
      v = v > 0.0f ? v : 0.0f;
      sH[(16 * wave + 8 * kh + j) * HSTR + t * 16 + m] = v;
    }
  }
  __syncthreads();

  // ---- GEMM 2: y = t @ W2 + b2 -------------------------------------------
  v8f acc2[4];
#pragma unroll
  for (int t = 0; t < 4; ++t) {
    float bv = b2[t * 16 + m];
    acc2[t] = (v8f){bv, bv, bv, bv, bv, bv, bv, bv};
  }
  for (int k0 = 0; k0 < DF; k0 += 4) {
    const int ka = k0 + 2 * kh;
    v2f a = *(const v2f*)&sH[arow * HSTR + ka];
#pragma unroll
    for (int t = 0; t < 4; ++t) {
      v2f b = *(const v2f*)&sW2[(ka >> 1) * WPSTR + (t * 16 + m) * 2];
      acc2[t] = __builtin_amdgcn_wmma_f32_16x16x4_f32(
          false, a, false, b, (short)0, acc2[t], false, false);
    }
  }

  // write y + accumulate BN statistics (per-column partials, masked on tail)
#pragma unroll
  for (int t = 0; t < 4; ++t) {
    float ssum = 0.0f, ssq = 0.0f;
#pragma unroll
    for (int j = 0; j < 8; ++j) {
      long gr = base + 16 * wave + 8 * kh + j;
      if (gr < n) {
        float v = acc2[t][j];
        y[gr * DF + t * 16 + m] = v;
        ssum += v;
        ssq += v * v;
      }
    }
    fatomic_add(&gsum[t * 16 + m], ssum);
    fatomic_add(&gsq[t * 16 + m], ssq);
  }
}

// --------------------------------------------------------- BN fold to scale/shift
__global__ __launch_bounds__(64) void bn_finalize_kernel(
    const float* __restrict__ gsum, const float* __restrict__ gsq,
    const float* __restrict__ gamma, const float* __restrict__ beta,
    float* __restrict__ scale, float* __restrict__ shift, int n) {
  int t = threadIdx.x;
  float invn = 1.0f / (float)n;
  float mean = gsum[t] * invn;
  float var = gsq[t] * invn - mean * mean;
  float rstd = rsqrtf(var + 1e-5f);
  float sc = gamma[t] * rstd;
  scale[t] = sc;
  shift[t] = beta[t] - mean * sc;
}

// --------------------------------------------------------- BN apply + ReLU (x4)
__global__ __launch_bounds__(256) void bn_relu_kernel(
    const float4* __restrict__ y, const float* __restrict__ scale,
    const float* __restrict__ shift, float4* __restrict__ xout, long total4) {
  long i = (long)blockIdx.x * blockDim.x + threadIdx.x;
  if (i >= total4) return;
  int d = (int)((i & 15) * 4);  // feature index of component .x (DF/4 == 16)
  float4 v = y[i];
  float4 r;
  r.x = fmaf(v.x, scale[d + 0], shift[d + 0]);
  r.y = fmaf(v.y, scale[d + 1], shift[d + 1]);
  r.z = fmaf(v.z, scale[d + 2], shift[d + 2]);
  r.w = fmaf(v.w, scale[d + 3], shift[d + 3]);
  r.x = r.x > 0.0f ? r.x : 0.0f;
  r.y = r.y > 0.0f ? r.y : 0.0f;
  r.z = r.z > 0.0f ? r.z : 0.0f;
  r.w = r.w > 0.0f ? r.w : 0.0f;
  xout[i] = r;
}

// --------------------------------------------------------- global mean pool (sums)
__global__ __launch_bounds__(256) void pool_kernel(
    const float* __restrict__ x, const int* __restrict__ batch,
    float* __restrict__ psum, float* __restrict__ pcnt, long total) {
  long i = (long)blockIdx.x * blockDim.x + threadIdx.x;
  if (i >= total) return;
  int node = (int)(i >> 6);
  int d = (int)(i & 63);
  int g = batch[node];
  fatomic_add(&psum[(long)g * DF + d], x[i]);
  if (d == 0) fatomic_add(&pcnt[g], 1.0f);
}

// --------------------------------------------------------- classifier head
__global__ __launch_bounds__(64) void classifier_kernel(
    const float* __restrict__ psum, const float* __restrict__ pcnt,
    const float* __restrict__ Wc, const float* __restrict__ bc,
    float* __restrict__ out) {
  __shared__ float r0[64];
  __shared__ float r1[64];
  int g = blockIdx.x;
  int t = threadIdx.x;
  float inv = 1.0f / fmaxf(pcnt[g], 1.0f);
  float p = psum[(long)g * DF + t] * inv;
  r0[t] = p * Wc[t * 2 + 0];
  r1[t] = p * Wc[t * 2 + 1];
  __syncthreads();
  for (int s = 32; s > 0; s >>= 1) {
    if (t < s) { r0[t] += r0[t + s]; r1[t] += r1[t + s]; }
    __syncthreads();
  }
  if (t == 0) {
    out[g * 2 + 0] = r0[0] + bc[0];
    out[g * 2 + 1] = r1[0] + bc[1];
  }
}

// ===========================================================================
extern "C" void kernel_launch(void* const* d_in, const int* in_sizes, int n_in,
                              void* d_out, int out_size, void* d_ws, size_t ws_size,
                              hipStream_t stream) {
  const float* x0    = (const float*)d_in[0];
  const int*   edge  = (const int*)d_in[1];
  const int*   batch = (const int*)d_in[2];
  const float* W1    = (const float*)d_in[3];
  const float* b1    = (const float*)d_in[4];
  const float* W2    = (const float*)d_in[5];
  const float* b2    = (const float*)d_in[6];
  const float* eps   = (const float*)d_in[7];
  const float* gamma = (const float*)d_in[8];
  const float* beta  = (const float*)d_in[9];
  const float* Wc    = (const float*)d_in[10];
  const float* bc    = (const float*)d_in[11];
  float* out = (float*)d_out;

  const int n = in_sizes[0] / DF;
  const int e = in_sizes[1] / 2;
  const int L = in_sizes[7];
  const int G = out_size / 2;
  const long ND = (long)n * DF;

  // workspace layout (floats)
  float* f     = (float*)d_ws;
  float* agg   = f;                     // ND
  float* gsum  = f + ND;                // 64
  float* gsq   = gsum + DF;             // 64
  float* scale = gsq + DF;              // 64
  float* shift = scale + DF;            // 64
  float* ybuf  = f + ND + 256;          // ND
  float* xA    = ybuf + ND;             // ND
  float* xB    = xA + ND;               // ND
  float* psum  = xB + ND;               // G*DF
  float* pcnt  = psum + (long)G * DF;   // G

  const int* src = edge;
  const int* dst = edge + e;

  const float* xin = x0;
  float* xout = xA;

  for (int i = 0; i < L; ++i) {
    long zcnt = ND + 128;  // agg + gsum + gsq are contiguous
    zero_kernel<<<(int)((zcnt + 255) / 256), 256, 0, stream>>>(agg, zcnt);

    long tot = (long)e * DF;
    scatter_kernel<<<(int)((tot + 255) / 256), 256, 0, stream>>>(xin, src, dst, agg, tot);

    int mblocks = (n + TILE_ROWS - 1) / TILE_ROWS;
    gin_mlp_kernel<<<mblocks, 256, 0, stream>>>(
        xin, agg,
        W1 + (long)i * DF * DF, b1 + (long)i * DF,
        W2 + (long)i * DF * DF, b2 + (long)i * DF,
        eps, i, ybuf, gsum, gsq, n);

    bn_finalize_kernel<<<1, DF, 0, stream>>>(gsum, gsq, gamma + (long)i * DF,
                                             beta + (long)i * DF, scale, shift, n);

    long t4 = ND / 4;
    bn_relu_kernel<<<(int)((t4 + 255) / 256), 256, 0, stream>>>(
        (const float4*)ybuf, scale, shift, (float4*)xout, t4);

    xin = xout;
    xout = (xout == xA) ? xB : xA;
  }

  long pz = (long)G * DF + G;
  zero_kernel<<<(int)((pz + 255) / 256), 256, 0, stream>>>(psum, pz);
  pool_kernel<<<(int)((ND + 255) / 256), 256, 0, stream>>>(xin, batch, psum, pcnt, ND);
  classifier_kernel<<<G, DF, 0, stream>>>(psum, pcnt, Wc, bc, out);
}